// GNO_layer_55259049230850
// MI455X (gfx1250) — compile-verified
//
#include <hip/hip_runtime.h>
#include <cstdint>

typedef __attribute__((ext_vector_type(2))) float v2f;
typedef __attribute__((ext_vector_type(8))) float v8f;

#define DD 64

// ---------------- zero out accumulator + degree ----------------
__global__ void zero_kernel(float* __restrict__ out, float* __restrict__ deg,
                            long n_out, long n_deg) {
    long i = (long)blockIdx.x * blockDim.x + threadIdx.x;
    if (i < n_out) out[i] = 0.0f;
    if (i < n_deg) deg[i] = 0.0f;
}

// ---------------- weighted in-degree on targets ----------------
__global__ void deg_kernel(const long long* __restrict__ ei, const float* __restrict__ ew,
                           float* __restrict__ deg, int E) {
    long e = (long)blockIdx.x * blockDim.x + threadIdx.x;
    if (e < E) atomicAdd(deg + ei[(long)E + e], ew[e]);
}

// self-loop contributes weight 1 -> deg+1 > 0 always (matches reference's where())
__global__ void dinv_kernel(const float* __restrict__ deg, float* __restrict__ dinv, int n) {
    int i = blockIdx.x * blockDim.x + threadIdx.x;
    if (i < n) dinv[i] = rsqrtf(deg[i] + 1.0f);
}

// ---------------- h = x @ W via V_WMMA_F32_16X16X4_F32 ----------------
// One wave: 16-row slab of x, all 64 output cols (4 col tiles, A-frag reused).
// A 16x4 f32 layout: lane m = lane&15; VGPR0/1 = K = klo / klo+1, klo = 2*(lane>=16).
// B 4x16 f32 layout: symmetric, per-column. C/D: vgpr i -> row i + 8*(lane>=16), col lane&15.
__global__ void gemm_kernel(const float* __restrict__ x, const float* __restrict__ W,
                            float* __restrict__ h, int n) {
    __shared__ float sW[DD * DD];
    int tid = threadIdx.x;
    for (int i = tid; i < DD * DD / 4; i += blockDim.x)
        ((float4*)sW)[i] = ((const float4*)W)[i];
    __syncthreads();

    int wave = tid >> 5;
    int lane = tid & 31;
    long row0 = ((long)blockIdx.x * 8 + wave) * 16;
    if (row0 >= n) return;  // whole-wave uniform exit; EXEC stays all-1 for WMMA

    int m   = lane & 15;
    int klo = (lane >> 4) * 2;

    v8f acc0 = {}, acc1 = {}, acc2 = {}, acc3 = {};
    const float* xrow = x + (row0 + m) * DD;

#pragma unroll
    for (int k0 = 0; k0 < DD; k0 += 4) {
        v2f a = *(const v2f*)(xrow + k0 + klo);           // 8B-aligned contiguous K pair
        const float* w0 = sW + (k0 + klo) * DD + m;       // row-major W[k][col]
        v2f b0; b0.x = w0[0];  b0.y = w0[DD];
        v2f b1; b1.x = w0[16]; b1.y = w0[DD + 16];
        v2f b2; b2.x = w0[32]; b2.y = w0[DD + 32];
        v2f b3; b3.x = w0[48]; b3.y = w0[DD + 48];
        acc0 = __builtin_amdgcn_wmma_f32_16x16x4_f32(false, a, false, b0, (short)0, acc0, false, false);
        acc1 = __builtin_amdgcn_wmma_f32_16x16x4_f32(false, a, false, b1, (short)0, acc1, false, false);
        acc2 = __builtin_amdgcn_wmma_f32_16x16x4_f32(false, a, false, b2, (short)0, acc2, false, false);
        acc3 = __builtin_amdgcn_wmma_f32_16x16x4_f32(false, a, false, b3, (short)0, acc3, false, false);
    }

    int mbase = (lane >> 4) * 8;
    float* hout = h + row0 * DD;
#pragma unroll
    for (int i = 0; i < 8; i++) {
        long r = (long)(mbase + i) * DD + m;
        hout[r +  0] = acc0[i];
        hout[r + 16] = acc1[i];
        hout[r + 32] = acc2[i];
        hout[r + 48] = acc3[i];
    }
}

// ---------------- edge gather + normalized scatter-add ----------------
// One wave per edge; lane handles dims 2l, 2l+1 (coalesced 256B gather per wave).
__global__ void scatter_kernel(const long long* __restrict__ ei, const float* __restrict__ ew,
                               const float* __restrict__ dinv, const float* __restrict__ h,
                               float* __restrict__ out, int E) {
    long e = (long)blockIdx.x * 8 + (threadIdx.x >> 5);
    if (e >= E) return;
    int lane = threadIdx.x & 31;
    long src = ei[e];
    long dst = ei[(long)E + e];
    float nrm = dinv[src] * ew[e] * dinv[dst];
    v2f hv = *(const v2f*)(h + src * DD + lane * 2);
    float* o = out + dst * DD + lane * 2;
    atomicAdd(o,     hv.x * nrm);
    atomicAdd(o + 1, hv.y * nrm);
}

// ---------------- self-loop + bias + ReLU ----------------
__global__ void finalize_kernel(const float* __restrict__ h, const float* __restrict__ dinv,
                                const float* __restrict__ bias, float* __restrict__ out,
                                long total) {
    long i = (long)blockIdx.x * blockDim.x + threadIdx.x;
    if (i >= total) return;
    long node = i >> 6;
    int  d    = (int)(i & 63);
    float di  = dinv[node];
    float v   = out[i] + h[i] * (di * di) + bias[d];
    out[i] = v > 0.0f ? v : 0.0f;
}

extern "C" void kernel_launch(void* const* d_in, const int* in_sizes, int n_in,
                              void* d_out, int out_size, void* d_ws, size_t ws_size,
                              hipStream_t stream) {
    const float*     x  = (const float*)d_in[0];
    const long long* ei = (const long long*)d_in[1];   // int64 edge_index [2,E] flat
    const float*     ew = (const float*)d_in[2];
    const float*     W  = (const float*)d_in[3];
    const float*     b  = (const float*)d_in[4];
    float* out = (float*)d_out;

    int N = in_sizes[0] / DD;
    int E = in_sizes[2];

    float* deg  = (float*)d_ws;        // [N]
    float* dinv = deg + N;             // [N]
    float* h    = dinv + N;            // [N*64]

    long total = (long)N * DD;
    int zgrid = (int)((total + 255) / 256);

    zero_kernel    <<<zgrid,              256, 0, stream>>>(out, deg, total, N);
    deg_kernel     <<<(E + 255) / 256,    256, 0, stream>>>(ei, ew, deg, E);
    dinv_kernel    <<<(N + 255) / 256,    256, 0, stream>>>(deg, dinv, N);
    int tiles = (N + 15) / 16;
    gemm_kernel    <<<(tiles + 7) / 8,    256, 0, stream>>>(x, W, h, N);
    scatter_kernel <<<(E + 7) / 8,        256, 0, stream>>>(ei, ew, dinv, h, out, E);
    finalize_kernel<<<zgrid,              256, 0, stream>>>(h, dinv, b, out, total);
}